// SNNEventModel_17059610100049
// MI455X (gfx1250) — compile-verified
//
#include <hip/hip_runtime.h>
#include <math.h>

typedef __attribute__((ext_vector_type(2))) float v2f;
typedef __attribute__((ext_vector_type(8))) float v8f;

#define BATCH 64
#define KTAP  8
#define LSEQ  16384
#define CHUNK 64
#define HALO  7

__device__ __forceinline__ v8f wmma_f32_16x16x4(v2f a, v2f b, v8f c) {
  // 8 args: (neg_a, A, neg_b, B, c_mod, C, reuse_a, reuse_b)
  return __builtin_amdgcn_wmma_f32_16x16x4_f32(false, a, false, b, (short)0, c,
                                               false, false);
}

__global__ __launch_bounds__(32) void snn_lif_kernel(
    const float* __restrict__ x,  const float* __restrict__ wa,
    const float* __restrict__ wb, const float* __restrict__ rta,
    const float* __restrict__ rtb, float* __restrict__ out)
{
  __shared__ float xs[2][CHUNK + HALO + 1];   // x window + causal halo
  __shared__ float wn_s[32][KTAP];            // normalized filter taps
  __shared__ float Is[32][CHUNK + 1];         // conv output, padded rows

  const int b  = blockIdx.x;
  const int l  = threadIdx.x;     // lane = output channel (0..31)
  const int g  = l >> 4;          // 0 = wa group, 1 = wb group
  const int fi = l & 15;

  // ---- per-filter L2-normalized weights -> LDS ----
  const float* wsrc = g ? wb : wa;
  float w[KTAP]; float ssq = 0.f;
  #pragma unroll
  for (int k = 0; k < KTAP; ++k) { float t = wsrc[fi * KTAP + k]; w[k] = t; ssq += t * t; }
  const float nrm = fmaxf(sqrtf(ssq), 1e-8f);
  #pragma unroll
  for (int k = 0; k < KTAP; ++k) wn_s[l][k] = w[k] / nrm;

  // ---- tau -> alpha ----
  const float rt  = (g ? rtb : rta)[fi];
  const float sp  = (rt > 20.f) ? rt : log1pf(expf(rt));  // softplus
  const float tau = sp + 1e-4f;
  const float alpha = expf(-1.0f / tau);
  const float oma   = 1.0f - alpha;

  __syncthreads();

  // ---- loop-invariant WMMA A operands (ISA 16x4 f32 A layout) ----
  const int m  = l & 15;
  const int ko = (l >= 16) ? 2 : 0;
  const v2f Aa0 = { wn_s[m][0 + ko],      wn_s[m][1 + ko] };
  const v2f Aa1 = { wn_s[m][4 + ko],      wn_s[m][5 + ko] };
  const v2f Ab0 = { wn_s[16 + m][0 + ko], wn_s[16 + m][1 + ko] };
  const v2f Ab1 = { wn_s[16 + m][4 + ko], wn_s[16 + m][5 + ko] };

  const size_t planeBL = (size_t)BATCH * 32 * LSEQ;
  float* vrow = out + ((size_t)(b * 32 + l)) * LSEQ;
  float* zrow = out + planeBL     + ((size_t)(b * 32 + l)) * LSEQ;
  float* srow = out + 2 * planeBL + ((size_t)(b * 32 + l)) * LSEQ;
  float* lrow = out + 3 * planeBL + (size_t)b * LSEQ;

  const float* xb = x + (size_t)b * 2 * LSEQ;

  float v = 0.f;

  for (int t0 = 0; t0 < LSEQ; t0 += CHUNK) {
    // ---- cooperative x window load (zeros left of t=0) ----
    for (int i = l; i < 2 * (CHUNK + HALO); i += 32) {
      int ch  = i / (CHUNK + HALO);
      int idx = i - ch * (CHUNK + HALO);
      int t   = t0 - HALO + idx;
      xs[ch][idx] = (t >= 0) ? xb[ch * LSEQ + t] : 0.f;
    }
    __syncthreads();

    // prefetch next chunk (global_prefetch_b8)
    if (t0 + CHUNK < LSEQ) {
      __builtin_prefetch(xb + t0 + CHUNK, 0, 0);
      __builtin_prefetch(xb + LSEQ + t0 + CHUNK, 0, 0);
    }

    // ---- causal conv via WMMA f32 16x16x4, 16-timestep subtiles ----
    #pragma unroll
    for (int tt = 0; tt < CHUNK; tt += 16) {
      const v2f Ba0 = { xs[0][tt + m + 0 + ko], xs[0][tt + m + 1 + ko] };
      const v2f Ba1 = { xs[0][tt + m + 4 + ko], xs[0][tt + m + 5 + ko] };
      const v2f Bb0 = { xs[1][tt + m + 0 + ko], xs[1][tt + m + 1 + ko] };
      const v2f Bb1 = { xs[1][tt + m + 4 + ko], xs[1][tt + m + 5 + ko] };
      v8f ca = {};
      ca = wmma_f32_16x16x4(Aa0, Ba0, ca);   // taps 0..3
      ca = wmma_f32_16x16x4(Aa1, Ba1, ca);   // taps 4..7
      v8f cb = {};
      cb = wmma_f32_16x16x4(Ab0, Bb0, cb);
      cb = wmma_f32_16x16x4(Ab1, Bb1, cb);
      const int rb = (l >= 16) ? 8 : 0;      // C layout: VGPR r -> M=r or r+8
      #pragma unroll
      for (int r = 0; r < 8; ++r) {
        Is[rb + r][tt + m]      = ca[r];
        Is[16 + rb + r][tt + m] = cb[r];
      }
    }
    __syncthreads();

    // ---- sequential LIF over the chunk; lane = channel ----
    float vv[4], zz[4], sv[4], lg[4];
    for (int tt = 0; tt < CHUNK; ++tt) {
      const float I     = Is[l][tt];
      const float v_pre = alpha * v + oma * I;
      const float z     = 15.0f * (v_pre - 0.25f);
      const float s     = (v_pre >= 0.25f) ? 1.0f : 0.0f;  // forward s == s_hard
      v = v_pre * (1.0f - s);

      const int q = tt & 3;
      vv[q] = v_pre; zz[q] = z; sv[q] = s;

      float zmax = z;                       // logits: max over 32 channels
      #pragma unroll
      for (int off = 16; off > 0; off >>= 1)
        zmax = fmaxf(zmax, __shfl_xor(zmax, off, 32));
      lg[q] = zmax;

      if (q == 3) {
        const int t = t0 + tt - 3;
        *(float4*)(vrow + t) = make_float4(vv[0], vv[1], vv[2], vv[3]);
        *(float4*)(zrow + t) = make_float4(zz[0], zz[1], zz[2], zz[3]);
        *(float4*)(srow + t) = make_float4(sv[0], sv[1], sv[2], sv[3]);
        if (l == 0)
          *(float4*)(lrow + t) = make_float4(lg[0], lg[1], lg[2], lg[3]);
      }
    }
    __syncthreads();   // protect xs/Is reuse next chunk
  }
}

extern "C" void kernel_launch(void* const* d_in, const int* in_sizes, int n_in,
                              void* d_out, int out_size, void* d_ws, size_t ws_size,
                              hipStream_t stream) {
  const float* x   = (const float*)d_in[0];
  const float* wa  = (const float*)d_in[1];
  const float* wb  = (const float*)d_in[2];
  const float* rta = (const float*)d_in[3];
  const float* rtb = (const float*)d_in[4];
  float* out = (float*)d_out;
  (void)in_sizes; (void)n_in; (void)out_size; (void)d_ws; (void)ws_size;

  snn_lif_kernel<<<BATCH, 32, 0, stream>>>(x, wa, wb, rta, rtb, out);
}